// WhisperDecoderStepSelfCached_58016418234553
// MI455X (gfx1250) — compile-verified
//
#include <hip/hip_runtime.h>
#include <math.h>

// ---------------- WMMA types ----------------
typedef __attribute__((ext_vector_type(2))) float v2f;
typedef __attribute__((ext_vector_type(8))) float v8f;

#define NLAYERS 12
#define BATCH   4
#define NHEADS  12
#define HDIM    64
#define DMODEL  768
#define FFNDIM  3072
#define VOCAB   51865
#define MAXDEC  448
#define ENCLEN  1500

__device__ __forceinline__ int imin(int a, int b) { return a < b ? a : b; }

// ---------------------------------------------------------------------------
// Generic batched GEMV via V_WMMA_F32_16X16X4_F32:
//   Y[4,N] = alpha * (X[4,K] @ W[N,K]^T)  (+ bias*bias_scale) (gelu?) (+ residual)
// One block = 4 waves; each wave owns a K/4 slice; LDS reduce; 16 outputs/block.
// Out-of-range A rows / B cols are CLAMPED (their C entries are never stored),
// so the inner loop is branch-free with EXEC all-ones, as WMMA requires.
// K must be a multiple of 64 (768 / 3072 here): manual 4x WMMA unroll.
// ---------------------------------------------------------------------------
__global__ __launch_bounds__(128)
void gemv_wmma_f32(const float* __restrict__ X, const float* __restrict__ W,
                   const float* __restrict__ bias, const float* __restrict__ residual,
                   float* __restrict__ Y, int K, int N,
                   float alpha, float bias_scale, int do_gelu)
{
    __shared__ float red[4][4][16];               // [wave][m][n]
    const int lane  = threadIdx.x & 31;
    const int wave  = threadIdx.x >> 5;
    const int n0    = blockIdx.x * 16;
    const int nlane = lane & 15;
    const int khalf = (lane >> 4) << 1;           // 0 or 2

    // Clamped rows: garbage C entries for clamped lanes are never stored.
    const float* wrow = W + (size_t)imin(n0 + nlane, N - 1) * K + khalf;
    const float* xrow = X + (size_t)imin(nlane, BATCH - 1) * K + khalf;

    const int kbeg = wave * (K >> 2);             // K/4 k-steps per wave slice
    const int kend = kbeg + (K >> 2);

    v8f c = {0.f, 0.f, 0.f, 0.f, 0.f, 0.f, 0.f, 0.f};
    for (int k = kbeg; k < kend; k += 16) {       // 4 WMMAs per iteration
        v2f a0, b0, a1, b1, a2, b2, a3, b3;
        a0.x = xrow[k];      a0.y = xrow[k + 1];
        b0.x = wrow[k];      b0.y = wrow[k + 1];
        a1.x = xrow[k + 4];  a1.y = xrow[k + 5];
        b1.x = wrow[k + 4];  b1.y = wrow[k + 5];
        a2.x = xrow[k + 8];  a2.y = xrow[k + 9];
        b2.x = wrow[k + 8];  b2.y = wrow[k + 9];
        a3.x = xrow[k + 12]; a3.y = xrow[k + 13];
        b3.x = wrow[k + 12]; b3.y = wrow[k + 13];
        c = __builtin_amdgcn_wmma_f32_16x16x4_f32(false, a0, false, b0, (short)0, c, false, false);
        c = __builtin_amdgcn_wmma_f32_16x16x4_f32(false, a1, false, b1, (short)0, c, false, false);
        c = __builtin_amdgcn_wmma_f32_16x16x4_f32(false, a2, false, b2, (short)0, c, false, false);
        c = __builtin_amdgcn_wmma_f32_16x16x4_f32(false, a3, false, b3, (short)0, c, false, false);
    }
    // lanes 0..15 hold M = j in c[j]; only batch rows 0..3 are valid.
    if (lane < 16) {
        red[wave][0][lane] = c[0];
        red[wave][1][lane] = c[1];
        red[wave][2][lane] = c[2];
        red[wave][3][lane] = c[3];
    }
    __syncthreads();
    const int idx = threadIdx.x;
    if (idx < 64) {
        const int mm = idx >> 4, nn = idx & 15;
        const int ncol = n0 + nn;
        if (ncol < N) {
            float s = red[0][mm][nn] + red[1][mm][nn] +
                      red[2][mm][nn] + red[3][mm][nn];
            s *= alpha;
            if (bias)     s += bias_scale * bias[ncol];
            if (do_gelu)  s  = 0.5f * s * (1.f + erff(s * 0.70710678118654752f));
            if (residual) s += residual[(size_t)mm * N + ncol];
            Y[(size_t)mm * N + ncol] = s;
        }
    }
}

// ---------------------------------------------------------------------------
// LayerNorm over last dim (D), one block per batch row.
// ---------------------------------------------------------------------------
__global__ __launch_bounds__(256)
void ln_kernel(const float* __restrict__ in, const float* __restrict__ gs,
               const float* __restrict__ gb, float* __restrict__ out, int D)
{
    __shared__ float rs[256], rq[256];
    const int b = blockIdx.x;
    const float* x = in + (size_t)b * D;
    float sum = 0.f, sq = 0.f;
    for (int j = threadIdx.x; j < D; j += 256) { float v = x[j]; sum += v; sq += v * v; }
    rs[threadIdx.x] = sum; rq[threadIdx.x] = sq;
    __syncthreads();
    for (int off = 128; off > 0; off >>= 1) {
        if (threadIdx.x < off) {
            rs[threadIdx.x] += rs[threadIdx.x + off];
            rq[threadIdx.x] += rq[threadIdx.x + off];
        }
        __syncthreads();
    }
    const float mean = rs[0] / (float)D;
    const float var  = rq[0] / (float)D - mean * mean;
    const float inv  = rsqrtf(var + 1e-5f);
    for (int j = threadIdx.x; j < D; j += 256)
        out[(size_t)b * D + j] = (x[j] - mean) * inv * gs[j] + gb[j];
}

// ---------------------------------------------------------------------------
// Embedding: h = token_emb[tok]*sqrt(D) + pos_emb[pos]
// ---------------------------------------------------------------------------
__global__ void embed_kernel(const int* __restrict__ tok, const int* __restrict__ pos,
                             const float* __restrict__ temb, const float* __restrict__ pemb,
                             float* __restrict__ h)
{
    const int b = blockIdx.x;
    const int t = tok[b], p = pos[b];
    for (int j = threadIdx.x; j < DMODEL; j += blockDim.x)
        h[(size_t)b * DMODEL + j] =
            temb[(size_t)t * DMODEL + j] * 27.712812921102035f +
            pemb[(size_t)p * DMODEL + j];
}

// ---------------------------------------------------------------------------
// Scatter new K/V rows into the (already cloned) output caches at pos[b].
// Caches layout per layer: [B, H, M, 64].  grid = B*H, block = 64.
// ---------------------------------------------------------------------------
__global__ void scatter_kv(const float* __restrict__ kin, const float* __restrict__ vin,
                           const int* __restrict__ pos, float* __restrict__ kc,
                           float* __restrict__ vc, int M)
{
    const int bh = blockIdx.x;
    const int b = bh / NHEADS, h = bh % NHEADS;
    const int p = pos[b];
    const int d = threadIdx.x;
    const size_t off = ((((size_t)b * NHEADS + h) * M) + p) * HDIM + d;
    kc[off] = kin[(size_t)b * DMODEL + h * HDIM + d];
    vc[off] = vin[(size_t)b * DMODEL + h * HDIM + d];
}

// ---------------------------------------------------------------------------
// Single-query attention per (b,h): scores over M keys, masked softmax,
// weighted V sum.  q is pre-scaled.  pos==nullptr => unmasked (cross-attn).
// grid = (H, B), block = 256.  K rows read as float4 (rows are 256B-strided).
// ---------------------------------------------------------------------------
__global__ __launch_bounds__(256)
void attn_kernel(const float* __restrict__ q, const float* __restrict__ kc,
                 const float* __restrict__ vc, float* __restrict__ out,
                 const int* __restrict__ pos, int M)
{
    __shared__ float4 sQ4[HDIM / 4];
    __shared__ float  sS[1504];
    __shared__ float  red[256];
    const int h = blockIdx.x, b = blockIdx.y;
    const int tid = threadIdx.x;

    if (tid < HDIM / 4)
        sQ4[tid] = ((const float4*)(q + (size_t)b * DMODEL + h * HDIM))[tid];
    __syncthreads();

    const int valid = pos ? (pos[b] + 1) : M;
    const float* kb = kc + (((size_t)b * NHEADS + h) * M) * HDIM;
    const float* vb = vc + (((size_t)b * NHEADS + h) * M) * HDIM;

    float lmax = -3.4e38f;
    for (int m2 = tid; m2 < M; m2 += 256) {
        float s = -3.4e38f;
        if (m2 < valid) {
            const float4* kr = (const float4*)(kb + (size_t)m2 * HDIM);
            s = 0.f;
#pragma unroll
            for (int d = 0; d < HDIM / 4; ++d) {
                const float4 kv = kr[d];
                const float4 qv = sQ4[d];
                s += qv.x * kv.x + qv.y * kv.y + qv.z * kv.z + qv.w * kv.w;
            }
        }
        sS[m2] = s;
        lmax = fmaxf(lmax, s);
    }
    red[tid] = lmax; __syncthreads();
    for (int off = 128; off > 0; off >>= 1) {
        if (tid < off) red[tid] = fmaxf(red[tid], red[tid + off]);
        __syncthreads();
    }
    const float mx = red[0];
    __syncthreads();

    float lsum = 0.f;
    for (int m2 = tid; m2 < M; m2 += 256) {
        const float e = expf(sS[m2] - mx);
        sS[m2] = e;
        lsum += e;
    }
    red[tid] = lsum; __syncthreads();
    for (int off = 128; off > 0; off >>= 1) {
        if (tid < off) red[tid] += red[tid + off];
        __syncthreads();
    }
    const float inv = 1.f / red[0];
    __syncthreads();

    const int d = tid & 63, g = tid >> 6;
    float acc = 0.f;
    for (int m2 = g; m2 < M; m2 += 4)
        acc += sS[m2] * vb[(size_t)m2 * HDIM + d];
    red[tid] = acc; __syncthreads();
    if (g == 0)
        out[(size_t)b * DMODEL + h * HDIM + d] =
            (red[d] + red[d + 64] + red[d + 128] + red[d + 192]) * inv;
}

// ---------------------------------------------------------------------------
extern "C" void kernel_launch(void* const* d_in, const int* in_sizes, int n_in,
                              void* d_out, int out_size, void* d_ws, size_t ws_size,
                              hipStream_t stream)
{
    const int*   tok   = (const int*)  d_in[0];
    const int*   pos   = (const int*)  d_in[1];
    const float* skc   = (const float*)d_in[2];
    const float* svc   = (const float*)d_in[3];
    const float* ckc   = (const float*)d_in[4];
    const float* cvc   = (const float*)d_in[5];
    const float* temb  = (const float*)d_in[6];
    const float* pemb  = (const float*)d_in[7];
    const float* qw    = (const float*)d_in[8];
    const float* qb    = (const float*)d_in[9];
    const float* kw    = (const float*)d_in[10];
    const float* vw    = (const float*)d_in[11];
    const float* vb    = (const float*)d_in[12];
    const float* ow    = (const float*)d_in[13];
    const float* ob    = (const float*)d_in[14];
    const float* cqw   = (const float*)d_in[15];
    const float* cqb   = (const float*)d_in[16];
    const float* cow   = (const float*)d_in[17];
    const float* cob   = (const float*)d_in[18];
    const float* ln1s  = (const float*)d_in[19];
    const float* ln1b  = (const float*)d_in[20];
    const float* ln2s  = (const float*)d_in[21];
    const float* ln2b  = (const float*)d_in[22];
    const float* ln3s  = (const float*)d_in[23];
    const float* ln3b  = (const float*)d_in[24];
    const float* fc1w  = (const float*)d_in[25];
    const float* fc1b  = (const float*)d_in[26];
    const float* fc2w  = (const float*)d_in[27];
    const float* fc2b  = (const float*)d_in[28];
    const float* lnfs  = (const float*)d_in[29];
    const float* lnfb  = (const float*)d_in[30];

    float* outp   = (float*)d_out;
    float* logits = outp;
    const size_t KVPL = (size_t)BATCH * NHEADS * MAXDEC * HDIM;   // per-layer self KV
    const size_t KV   = (size_t)NLAYERS * KVPL;                   // 16,515,072
    float* okc = outp + (size_t)BATCH * VOCAB;
    float* ovc = okc + KV;

    // Clone caches into outputs (overlaps with compute in the graph).
    hipMemcpyAsync(okc, skc, KV * sizeof(float), hipMemcpyDeviceToDevice, stream);
    hipMemcpyAsync(ovc, svc, KV * sizeof(float), hipMemcpyDeviceToDevice, stream);

    // Workspace layout (floats): h, x, q, k, v, a (4x768 each), f (4x3072)
    float* ws = (float*)d_ws;
    float* hbuf = ws;
    float* xbuf = hbuf + BATCH * DMODEL;
    float* qbuf = xbuf + BATCH * DMODEL;
    float* kbuf = qbuf + BATCH * DMODEL;
    float* vbuf = kbuf + BATCH * DMODEL;
    float* abuf = vbuf + BATCH * DMODEL;
    float* fbuf = abuf + BATCH * DMODEL;

    embed_kernel<<<BATCH, 256, 0, stream>>>(tok, pos, temb, pemb, hbuf);

    const float scaling = 0.125f;   // HEAD_DIM^-0.5
    const int NT  = DMODEL / 16;    // 48 tiles
    const int NTF = FFNDIM / 16;    // 192 tiles

    for (int l = 0; l < NLAYERS; ++l) {
        const size_t wo = (size_t)l * DMODEL * DMODEL;
        // ---- self attention ----
        ln_kernel<<<BATCH, 256, 0, stream>>>(hbuf, ln1s + l * DMODEL, ln1b + l * DMODEL, xbuf, DMODEL);
        gemv_wmma_f32<<<NT, 128, 0, stream>>>(xbuf, qw + wo, qb + l * DMODEL, nullptr,
                                              qbuf, DMODEL, DMODEL, scaling, scaling, 0);
        gemv_wmma_f32<<<NT, 128, 0, stream>>>(xbuf, kw + wo, nullptr, nullptr,
                                              kbuf, DMODEL, DMODEL, 1.f, 0.f, 0);
        gemv_wmma_f32<<<NT, 128, 0, stream>>>(xbuf, vw + wo, vb + l * DMODEL, nullptr,
                                              vbuf, DMODEL, DMODEL, 1.f, 1.f, 0);
        float* lkc = okc + (size_t)l * KVPL;
        float* lvc = ovc + (size_t)l * KVPL;
        scatter_kv<<<BATCH * NHEADS, HDIM, 0, stream>>>(kbuf, vbuf, pos, lkc, lvc, MAXDEC);
        attn_kernel<<<dim3(NHEADS, BATCH), 256, 0, stream>>>(qbuf, lkc, lvc, abuf, pos, MAXDEC);
        gemv_wmma_f32<<<NT, 128, 0, stream>>>(abuf, ow + wo, ob + l * DMODEL, hbuf,
                                              hbuf, DMODEL, DMODEL, 1.f, 1.f, 0);
        // ---- cross attention ----
        ln_kernel<<<BATCH, 256, 0, stream>>>(hbuf, ln2s + l * DMODEL, ln2b + l * DMODEL, xbuf, DMODEL);
        gemv_wmma_f32<<<NT, 128, 0, stream>>>(xbuf, cqw + wo, cqb + l * DMODEL, nullptr,
                                              qbuf, DMODEL, DMODEL, scaling, scaling, 0);
        attn_kernel<<<dim3(NHEADS, BATCH), 256, 0, stream>>>(
            qbuf,
            ckc + (size_t)l * BATCH * NHEADS * ENCLEN * HDIM,
            cvc + (size_t)l * BATCH * NHEADS * ENCLEN * HDIM,
            abuf, nullptr, ENCLEN);
        gemv_wmma_f32<<<NT, 128, 0, stream>>>(abuf, cow + wo, cob + l * DMODEL, hbuf,
                                              hbuf, DMODEL, DMODEL, 1.f, 1.f, 0);
        // ---- FFN ----
        ln_kernel<<<BATCH, 256, 0, stream>>>(hbuf, ln3s + l * DMODEL, ln3b + l * DMODEL, xbuf, DMODEL);
        gemv_wmma_f32<<<NTF, 128, 0, stream>>>(xbuf, fc1w + (size_t)l * FFNDIM * DMODEL,
                                               fc1b + l * FFNDIM, nullptr,
                                               fbuf, DMODEL, FFNDIM, 1.f, 1.f, 1);
        gemv_wmma_f32<<<NT, 128, 0, stream>>>(fbuf, fc2w + (size_t)l * DMODEL * FFNDIM,
                                              fc2b + l * DMODEL, hbuf,
                                              hbuf, FFNDIM, DMODEL, 1.f, 1.f, 0);
    }
    // ---- final LN + tied-embedding logits ----
    ln_kernel<<<BATCH, 256, 0, stream>>>(hbuf, lnfs, lnfb, xbuf, DMODEL);
    gemv_wmma_f32<<<(VOCAB + 15) / 16, 128, 0, stream>>>(xbuf, temb, nullptr, nullptr,
                                                         logits, DMODEL, VOCAB, 1.f, 0.f, 0);
}